// LatticeSnake_47253230190946
// MI455X (gfx1250) — compile-verified
//
#include <hip/hip_runtime.h>
#include <stdint.h>

// LatticeSnake: B=32, L=512, W=9 -> out [32,512,9,9,9] f32 (47.8 MB, write-bound).
// One wave32 per residue window, LDS accumulation with ds_add_f32,
// TDM (tensor_load_to_lds / tensor_store_from_lds) for all bulk data movement.

#define B_      32
#define L_      512
#define W_      9
#define W3_     729
#define HALF_   4
#define WPB     8        // windows (waves) per block
#define THREADS 256      // 8 wave32s

typedef uint32_t u32x4 __attribute__((ext_vector_type(4)));
typedef int32_t  i32x8 __attribute__((ext_vector_type(8)));
typedef int32_t  i32x4 __attribute__((ext_vector_type(4)));

// Low 32 bits of a shared-aperture flat address == LDS byte offset (ISA 10.2).
static __device__ __forceinline__ uint32_t lds_off(const void* p) {
    return (uint32_t)(uintptr_t)p;
}

// Build a 1-D TDM descriptor: nDwords 4-byte elements, global <-> LDS.
// Group0: count=1 | lds_addr | global_addr[56:0] | type=2.
// Group1: data_size=4B, tensor_dim0=tile_dim0=nDwords, tensor_dim1=tile_dim1=1,
//         tensor_dim0_stride=nDwords, no padding/iterate/multicast.
static __device__ __forceinline__ void tdm_desc(u32x4& g0, i32x8& g1,
                                                const void* gptr,
                                                uint32_t ldsByteOff,
                                                uint32_t nDwords) {
    uint64_t ga = (uint64_t)(uintptr_t)gptr;
    g0 = (u32x4){ 1u,                                   // count=1, user D#
                  ldsByteOff,                           // lds_addr (bytes)
                  (uint32_t)ga,                         // global_addr[31:0]
                  (uint32_t)(ga >> 32) | (2u << 30) };  // addr[56:32] | type=2
    g1 = (i32x8){ (int)(2u << 16),        // dw0: data_size=2 (4 bytes), mask=0
                  (int)(nDwords << 16),   // dw1: tensor_dim0[15:0] in [31:16]
                  (int)(1u << 16),        // dw2: tensor_dim0 hi=0 | tensor_dim1=1
                  (int)(nDwords << 16),   // dw3: tensor_dim1 hi=0 | tile_dim0
                  1,                      // dw4: tile_dim1=1, tile_dim2=0
                  (int)nDwords,           // dw5: tensor_dim0_stride[31:0]
                  0,                      // dw6: stride hi | tensor_dim1_stride lo
                  0 };                    // dw7: tensor_dim1_stride hi
}

__global__ __launch_bounds__(THREADS)
void LatticeSnake_kernel(const float* __restrict__ acids,
                         const float* __restrict__ mask,
                         const int*   __restrict__ idx,
                         float*       __restrict__ out) {
    __shared__ int   rawIdx[L_ * 3];     //  6144 B : idx[b] staged by TDM
    __shared__ float rawA[L_];           //  2048 B : acids[b]
    __shared__ float rawM[L_];           //  2048 B : mask[b]
    __shared__ int   cx[1024];           //  4096 B : snake point coords (doubled)
    __shared__ int   cy[1024];
    __shared__ int   cz[1024];
    __shared__ float cv[1024];           //  4096 B : masked point values
    __shared__ float acc[WPB * W3_];     // 23328 B : 8 windows of 9^3 floats

    const int b = blockIdx.y;
    const int t = threadIdx.x;

    // --- Stage this batch's raw inputs into LDS via the Tensor Data Mover ---
    if (t == 0) {
        u32x4 g0; i32x8 g1;
        i32x4 z4 = (i32x4){0, 0, 0, 0};
        i32x8 z8 = (i32x8){0, 0, 0, 0, 0, 0, 0, 0};
        tdm_desc(g0, g1, idx   + (size_t)b * (L_ * 3), lds_off(rawIdx), L_ * 3);
        __builtin_amdgcn_tensor_load_to_lds(g0, g1, z4, z4, z8, 0);
        tdm_desc(g0, g1, acids + (size_t)b * L_,       lds_off(rawA),   L_);
        __builtin_amdgcn_tensor_load_to_lds(g0, g1, z4, z4, z8, 0);
        tdm_desc(g0, g1, mask  + (size_t)b * L_,       lds_off(rawM),   L_);
        __builtin_amdgcn_tensor_load_to_lds(g0, g1, z4, z4, z8, 0);
        __builtin_amdgcn_s_wait_tensorcnt(0);
    }
    __syncthreads();

    // --- Derive the M=1023 snake points (residues + bond midpoints) ---
    for (int j = t; j < L_; j += THREADS) {
        cx[j] = 2 * (rawIdx[3 * j + 0] + (L_ - 1));
        cy[j] = 2 * (rawIdx[3 * j + 1] + (L_ - 1));
        cz[j] = 2 * (rawIdx[3 * j + 2] + (L_ - 1));
        cv[j] = rawA[j] * rawM[j];
    }
    for (int j = t; j < L_ - 1; j += THREADS) {
        cx[L_ + j] = rawIdx[3 * j + 0] + rawIdx[3 * j + 3] + 2 * (L_ - 1);
        cy[L_ + j] = rawIdx[3 * j + 1] + rawIdx[3 * j + 4] + 2 * (L_ - 1);
        cz[L_ + j] = rawIdx[3 * j + 2] + rawIdx[3 * j + 5] + 2 * (L_ - 1);
        cv[L_ + j] = (rawA[j] + rawA[j + 1] + 1.0f) * rawM[j + 1];
    }
    if (t == 0) {   // pad slot 1023: coordinates that can never be in-window
        cx[1023] = 1 << 20; cy[1023] = 1 << 20; cz[1023] = 1 << 20;
        cv[1023] = 0.0f;
    }
    for (int k = t; k < WPB * W3_; k += THREADS) acc[k] = 0.0f;
    __syncthreads();

    // --- One wave per window: sweep all points, scatter into LDS window ---
    const int wave = t >> 5;          // wave32
    const int lane = t & 31;
    const int i    = blockIdx.x * WPB + wave;      // residue index (center)
    const int ccx  = cx[i], ccy = cy[i], ccz = cz[i];
    float* accw = &acc[wave * W3_];

#pragma unroll 4
    for (int j = lane; j < 1024; j += 32) {
        int rx = cx[j] - ccx + HALF_;
        int ry = cy[j] - ccy + HALF_;
        int rz = cz[j] - ccz + HALF_;
        if (((unsigned)rx < (unsigned)W_) &
            ((unsigned)ry < (unsigned)W_) &
            ((unsigned)rz < (unsigned)W_)) {
            // walk may revisit a cell -> LDS float atomic (ds_add_f32)
            atomicAdd(&accw[(rx * W_ + ry) * W_ + rz], cv[j]);
        }
    }
    __syncthreads();

    // --- Stream the 8 finished windows (contiguous 23 KB) out via TDM ---
    if (t == 0) {
        u32x4 g0; i32x8 g1;
        i32x4 z4 = (i32x4){0, 0, 0, 0};
        i32x8 z8 = (i32x8){0, 0, 0, 0, 0, 0, 0, 0};
        float* dst = out + ((size_t)b * L_ + (size_t)blockIdx.x * WPB) * W3_;
        tdm_desc(g0, g1, dst, lds_off(acc), WPB * W3_);
        __builtin_amdgcn_tensor_store_from_lds(g0, g1, z4, z4, z8, 0);
        __builtin_amdgcn_s_wait_tensorcnt(0);
    }
}

extern "C" void kernel_launch(void* const* d_in, const int* in_sizes, int n_in,
                              void* d_out, int out_size, void* d_ws, size_t ws_size,
                              hipStream_t stream) {
    (void)in_sizes; (void)n_in; (void)out_size; (void)d_ws; (void)ws_size;
    const float* acids = (const float*)d_in[0];
    const float* mask  = (const float*)d_in[1];
    const int*   idx   = (const int*)d_in[2];
    float*       out   = (float*)d_out;   // [32,512,9,9,9] f32

    dim3 grid(L_ / WPB, B_);              // (64, 32)
    LatticeSnake_kernel<<<grid, THREADS, 0, stream>>>(acids, mask, idx, out);
}